// SmoothLoss_76175539962304
// MI455X (gfx1250) — compile-verified
//
#include <hip/hip_runtime.h>
#include <math.h>

typedef __attribute__((ext_vector_type(2))) float v2f;
typedef __attribute__((ext_vector_type(8))) float v8f;

#define BATCH 8
#define NPTS 4096
#define ROWS_PER_WAVE 16
#define WAVES_PER_BLOCK 8
#define ROWS_PER_BLOCK (ROWS_PER_WAVE * WAVES_PER_BLOCK)   // 128

// Compiler-only ordering fence: prevents IR-level reordering of the LDS
// write->read transpose. No instruction emitted; the HW keeps same-wave DS
// ops in order (CDNA5 ISA: LDS ops in-order within a wave), and the compiler
// inserts s_wait_dscnt for the load-result register hazards.
#define LDS_ORDER_FENCE() asm volatile("" ::: "memory")

__device__ __forceinline__ void top5_insert(float v, float& m0, float& m1,
                                            float& m2, float& m3, float& m4) {
  // Maintain m0<=m1<=m2<=m3<=m4 = five smallest seen. Branchless bubble pass.
  float t = v, u;
  u = fmaxf(m0, t); m0 = fminf(m0, t); t = u;
  u = fmaxf(m1, t); m1 = fminf(m1, t); t = u;
  u = fmaxf(m2, t); m2 = fminf(m2, t); t = u;
  u = fmaxf(m3, t); m3 = fminf(m3, t); t = u;
  m4 = fminf(m4, t);
}

__global__ void init_ws_kernel(double* ws) {
  int i = threadIdx.x;
  if (i < 2 * BATCH) ws[i] = 0.0;
}

__global__ __launch_bounds__(256) void knn_loss_kernel(const float* __restrict__ x,
                                                       double* __restrict__ ws) {
  const int lane = threadIdx.x & 31;
  const int wave = threadIdx.x >> 5;
  const int hi   = lane >> 4;     // 0 = lanes 0-15, 1 = lanes 16-31
  const int l4   = lane & 15;

  const int blocks_per_batch = NPTS / ROWS_PER_BLOCK;     // 32
  const int b    = blockIdx.x / blocks_per_batch;
  const int row0 = (blockIdx.x % blocks_per_batch) * ROWS_PER_BLOCK
                 + wave * ROWS_PER_WAVE;

  const float* xb = x + (size_t)b * NPTS * 3;

  // Per-wave private 16x16 tile (padded): waves never share -> no barriers.
  __shared__ float lds_tile[WAVES_PER_BLOCK][16][17];

  // ---- A tile: 16 rows x K=4 (x,y,z,0). f32 16x4 A layout:
  // lanes 0-15: VGPR0=K0, VGPR1=K1 ; lanes 16-31: VGPR0=K2, VGPR1=K3
  const int arow = row0 + l4;               // both halves load rows 0..15
  const float ax = xb[arow * 3 + 0];
  const float ay = xb[arow * 3 + 1];
  const float az = xb[arow * 3 + 2];
  v2f A;
  A.x = hi ? az : ax;
  A.y = hi ? 0.0f : ay;
  const float sq_row = ax * ax + ay * ay + az * az;   // |x_row(l4)|^2, in every lane

  // C/D layout: VGPR v -> row M = v + 8*hi, col N = l4. Broadcast row norms
  // via lane shuffles instead of LDS (lane 8*hi+v holds row 8*hi+v's norm).
  float sqr[8];
#pragma unroll
  for (int v = 0; v < 8; ++v) sqr[v] = __shfl(sq_row, 8 * hi + v, 32);

  const float INF = __builtin_huge_valf();
  float m0 = INF, m1 = INF, m2 = INF, m3 = INF, m4 = INF;

  for (int j0 = 0; j0 < NPTS; j0 += 16) {
    const int jc = j0 + l4;
    const float bx = xb[jc * 3 + 0];
    const float by = xb[jc * 3 + 1];
    const float bz = xb[jc * 3 + 2];
    const float sqj = bx * bx + by * by + bz * bz;
    // B tile 4x16: lanes 0-15: VGPR0=K0,VGPR1=K1 ; lanes 16-31: VGPR0=K2,VGPR1=K3
    v2f B;
    B.x = hi ? bz : bx;
    B.y = hi ? 0.0f : by;

    v8f C = {};
    C = __builtin_amdgcn_wmma_f32_16x16x4_f32(false, A, false, B,
                                              (short)0, C, false, false);

    // d2 = |xi|^2 + |xj|^2 - 2 <xi,xj>, clamped at 0; transpose via LDS.
#pragma unroll
    for (int v = 0; v < 8; ++v) {
      float d2 = fmaxf(sqr[v] + sqj - 2.0f * C[v], 0.0f);
      lds_tile[wave][8 * hi + v][l4] = d2;
    }
    LDS_ORDER_FENCE();   // same-wave DS ops are in-order; just stop IR reordering
    // Row-wise top-5: lane handles row l4, columns [8*hi, 8*hi+8).
#pragma unroll
    for (int c = 0; c < 8; ++c) {
      top5_insert(lds_tile[wave][l4][8 * hi + c], m0, m1, m2, m3, m4);
    }
    LDS_ORDER_FENCE();   // keep next iteration's stores after these loads
  }

  // Merge the hi-half lane's top-5 into the lo-half lane (same row).
  float o0 = __shfl(m0, lane + 16, 32);
  float o1 = __shfl(m1, lane + 16, 32);
  float o2 = __shfl(m2, lane + 16, 32);
  float o3 = __shfl(m3, lane + 16, 32);
  float o4 = __shfl(m4, lane + 16, 32);
  top5_insert(o0, m0, m1, m2, m3, m4);
  top5_insert(o1, m0, m1, m2, m3, m4);
  top5_insert(o2, m0, m1, m2, m3, m4);
  top5_insert(o3, m0, m1, m2, m3, m4);
  top5_insert(o4, m0, m1, m2, m3, m4);

  // m0 ~ 0 is the self-distance; mean of sqrt of the next 4.
  float pp = 0.0f;
  if (hi == 0) pp = 0.25f * (sqrtf(m1) + sqrtf(m2) + sqrtf(m3) + sqrtf(m4));
  float pp2 = pp * pp;
#pragma unroll
  for (int off = 16; off >= 1; off >>= 1) {
    pp  += __shfl_xor(pp,  off, 32);
    pp2 += __shfl_xor(pp2, off, 32);
  }
  if (lane == 0) {
    atomicAdd(&ws[b],         (double)pp);
    atomicAdd(&ws[BATCH + b], (double)pp2);
  }
}

__global__ void finalize_kernel(const double* __restrict__ ws,
                                float* __restrict__ out) {
  if (threadIdx.x == 0 && blockIdx.x == 0) {
    double acc = 0.0;
    for (int b = 0; b < BATCH; ++b) {
      double s    = ws[b];
      double s2   = ws[BATCH + b];
      double mean = s / (double)NPTS;
      double var  = (s2 - (double)NPTS * mean * mean) / (double)(NPTS - 1);
      if (var < 0.0) var = 0.0;
      acc += sqrt(var);
    }
    out[0] = (float)(acc / (double)BATCH);
  }
}

extern "C" void kernel_launch(void* const* d_in, const int* in_sizes, int n_in,
                              void* d_out, int out_size, void* d_ws, size_t ws_size,
                              hipStream_t stream) {
  (void)in_sizes; (void)n_in; (void)out_size; (void)ws_size;
  const float* x = (const float*)d_in[0];
  double* ws = (double*)d_ws;
  float* out = (float*)d_out;

  init_ws_kernel<<<1, 32, 0, stream>>>(ws);
  const int grid = BATCH * (NPTS / ROWS_PER_BLOCK);   // 256 blocks, 8 waves each
  knn_loss_kernel<<<grid, 256, 0, stream>>>(x, ws);
  finalize_kernel<<<1, 32, 0, stream>>>(ws, out);
}